// BipartiteHeteroGNN_13718125543889
// MI455X (gfx1250) — compile-verified
//
#include <hip/hip_runtime.h>

namespace {

constexpr int HIDC = 128;
constexpr int NC   = 20000;
constexpr int NV   = 50000;
constexpr int NE   = 500000;
constexpr int NLAY = 3;

typedef float v2f __attribute__((ext_vector_type(2)));
typedef float v8f __attribute__((ext_vector_type(8)));

#define GEMM_RELU 1
#define GEMM_ADD  2

#if defined(__HIP_DEVICE_COMPILE__) && __has_builtin(__builtin_amdgcn_wmma_f32_16x16x4_f32)
#define USE_WMMA_F32 1
#else
#define USE_WMMA_F32 0
#endif

// ---------------------------------------------------------------------------
// Pack W[128][128] (row-major, W[k][n]) into B-fragment order for
// V_WMMA_F32_16X16X4_F32:
//   pair index i = ((t*8 + j)*32 + L),  t = k-step (k=4t), j = 16-col tile, L = lane
//   lane L<16  -> (W[4t+0][16j+L],    W[4t+1][16j+L])      (VGPR0=K0, VGPR1=K1)
//   lane L>=16 -> (W[4t+2][16j+L-16], W[4t+3][16j+L-16])   (VGPR0=K2, VGPR1=K3)
// ---------------------------------------------------------------------------
__global__ __launch_bounds__(256) void pack_w_kernel(const float* __restrict__ W,
                                                     float2* __restrict__ Wpk) {
  int i = blockIdx.x * blockDim.x + threadIdx.x;     // 0..8191
  if (i >= 32 * 8 * 32) return;
  int L = i & 31;
  int j = (i >> 5) & 7;
  int t = i >> 8;
  int k = 4 * t + ((L < 16) ? 0 : 2);
  int n = 16 * j + (L & 15);
  float2 p;
  p.x = W[k * HIDC + n];
  p.y = W[(k + 1) * HIDC + n];
  Wpk[i] = p;
}

// ---------------------------------------------------------------------------
// OUT[n,128] = op( Z[n,128] @ W[128,128] + bias )   via fp32 WMMA 16x16x4.
// One wave per 16-row stripe; wave computes the full 16x128 output
// (8 accumulator tiles). Packed W is staged in LDS (64 KB) per workgroup.
// Safe for OUT == Z (each wave reads only its own rows, writes at the end).
// Requires nrows % 16 == 0 (true for 20000 / 50000).
// ---------------------------------------------------------------------------
__global__ __launch_bounds__(256) void gemm128_kernel(const float* __restrict__ Z,
                                                      const float* __restrict__ Wpk,
                                                      const float* __restrict__ bias,
                                                      float* __restrict__ OUT,
                                                      int nrows, int flags) {
  __shared__ float4 smem[4096];                      // 64 KB packed-W stage
  const float4* wsrc = (const float4*)Wpk;
  for (int i = threadIdx.x; i < 4096; i += 256) smem[i] = wsrc[i];
  __syncthreads();
  const float2* wl = (const float2*)smem;

  const int wave = (int)((blockIdx.x * blockDim.x + threadIdx.x) >> 5);
  const int lane = (int)(threadIdx.x & 31);
  const int m0 = wave * 16;
  if (m0 >= nrows) return;                           // wave-uniform: EXEC stays all-1s

  v8f acc[8];

  // A fragment addressing (ISA 7.12.2, 32-bit A 16x4):
  //  lanes 0-15 : row m0+L,    VGPR0=K0, VGPR1=K1
  //  lanes16-31 : row m0+L-16, VGPR0=K2, VGPR1=K3
  const int rowA = m0 + (lane & 15);
  const int koff = (lane >> 4) << 1;                 // 0 or 2
  const float* zrow = Z + (size_t)rowA * HIDC + koff;

#if USE_WMMA_F32
  const v8f vzero = {0.f, 0.f, 0.f, 0.f, 0.f, 0.f, 0.f, 0.f};
#pragma unroll
  for (int j = 0; j < 8; ++j) acc[j] = vzero;

#pragma unroll 4
  for (int t = 0; t < 32; ++t) {
    float2 av = *(const float2*)(zrow + 4 * t);
    v2f a; a[0] = av.x; a[1] = av.y;
#pragma unroll
    for (int j = 0; j < 8; ++j) {
      float2 bv = wl[t * 256 + j * 32 + lane];       // conflict-free ds_load_b64
      v2f b; b[0] = bv.x; b[1] = bv.y;
      acc[j] = __builtin_amdgcn_wmma_f32_16x16x4_f32(
          /*neg_a=*/false, a, /*neg_b=*/false, b,
          /*c_mod=*/(short)0, acc[j], /*reuse_a=*/false, /*reuse_b=*/false);
    }
  }
#else
  // Scalar fallback (kept only so the file compiles if the builtin guess is wrong)
  for (int j = 0; j < 8; ++j) {
    for (int r = 0; r < 8; ++r) {
      float s = 0.f;
      int row = m0 + ((lane >> 4) * 8) + r;
      for (int k = 0; k < HIDC; ++k) {
        int pj = ((k >> 2) * 8 + j) * 32 + (lane & 15) + (((k & 3) >> 1) << 4);
        float2 wv2 = wl[pj];
        float wv = (k & 1) ? wv2.y : wv2.x;
        s += Z[(size_t)row * HIDC + k] * wv;
      }
      acc[j][r] = s;
    }
  }
#endif

  // D layout (ISA 7.12.2, 32-bit C/D 16x16):
  //  lanes 0-15 : VGPR r -> (row m0+r,   col 16j+L)
  //  lanes16-31 : VGPR r -> (row m0+8+r, col 16j+L-16)
  const int ncl   = lane & 15;
  const int rbase = (lane >> 4) * 8;
#pragma unroll
  for (int j = 0; j < 8; ++j) {
    const int n = j * 16 + ncl;
    const float bv = bias[n];
#pragma unroll
    for (int r = 0; r < 8; ++r) {
      float v = acc[j][r] + bv;
      float* p = OUT + (size_t)(m0 + rbase + r) * HIDC + n;
      if (flags & GEMM_ADD) v += *p;
      if (flags & GEMM_RELU) v = fmaxf(v, 0.f);
      *p = v;
    }
  }
}

__global__ __launch_bounds__(256) void zero_kernel(float4* __restrict__ p, int n4) {
  int i = blockIdx.x * blockDim.x + threadIdx.x;
  if (i < n4) p[i] = make_float4(0.f, 0.f, 0.f, 0.f);
}

// out[i][:] = relu(s[i] * W1[0][:] + b1[:])   (fin==1 first linear of the encoders)
__global__ __launch_bounds__(256) void outer_relu_kernel(const float* __restrict__ s,
    const float* __restrict__ W1, const float* __restrict__ b1,
    float* __restrict__ out, int n) {
  int wid  = (int)((blockIdx.x * blockDim.x + threadIdx.x) >> 5);
  int lane = (int)(threadIdx.x & 31);
  if (wid >= n) return;
  float sv = s[wid];
  float4 w  = ((const float4*)W1)[lane];
  float4 bb = ((const float4*)b1)[lane];
  float4 r;
  r.x = fmaxf(fmaf(sv, w.x, bb.x), 0.f);
  r.y = fmaxf(fmaf(sv, w.y, bb.y), 0.f);
  r.z = fmaxf(fmaf(sv, w.z, bb.z), 0.f);
  r.w = fmaxf(fmaf(sv, w.w, bb.w), 0.f);
  ((float4*)out)[(size_t)wid * 32 + lane] = r;
}

// One wave per edge: msg = relu(x_src[src] + attr*eW + eb); atomicAdd into aggr[dst].
__global__ __launch_bounds__(256) void edge_msg_kernel(const float* __restrict__ xsrc,
    const int* __restrict__ ei, const float* __restrict__ attr,
    const float* __restrict__ eW, const float* __restrict__ eb,
    float* __restrict__ aggr, int ne) {
  int e    = (int)((blockIdx.x * blockDim.x + threadIdx.x) >> 5);
  int lane = (int)(threadIdx.x & 31);
  if (e >= ne) return;
  int s = ei[e];
  int d = ei[ne + e];
  float a = attr[e];
  float4 w  = ((const float4*)eW)[lane];
  float4 bb = ((const float4*)eb)[lane];
  float4 x  = ((const float4*)xsrc)[(size_t)s * 32 + lane];
  float4 m;
  m.x = fmaxf(x.x + fmaf(a, w.x, bb.x), 0.f);
  m.y = fmaxf(x.y + fmaf(a, w.y, bb.y), 0.f);
  m.z = fmaxf(x.z + fmaf(a, w.z, bb.z), 0.f);
  m.w = fmaxf(x.w + fmaf(a, w.w, bb.w), 0.f);
  float* dst = aggr + (size_t)d * HIDC + lane * 4;
  atomicAdd(dst + 0, m.x);
  atomicAdd(dst + 1, m.y);
  atomicAdd(dst + 2, m.z);
  atomicAdd(dst + 3, m.w);
}

// aggr = (1+eps)*x_dst + aggr   (in place)
__global__ __launch_bounds__(256) void gine_pre_kernel(float* __restrict__ aggr,
    const float* __restrict__ xdst, const float* __restrict__ eps, int n4) {
  int i = blockIdx.x * blockDim.x + threadIdx.x;
  if (i >= n4) return;
  float k = 1.f + eps[0];
  float4 a = ((const float4*)aggr)[i];
  float4 x = ((const float4*)xdst)[i];
  a.x = fmaf(k, x.x, a.x);
  a.y = fmaf(k, x.y, a.y);
  a.z = fmaf(k, x.z, a.z);
  a.w = fmaf(k, x.w, a.w);
  ((float4*)aggr)[i] = a;
}

__global__ __launch_bounds__(256) void relu_copy_kernel(const float* __restrict__ h,
    float* __restrict__ x, int n4) {
  int i = blockIdx.x * blockDim.x + threadIdx.x;
  if (i >= n4) return;
  float4 v = ((const float4*)h)[i];
  v.x = fmaxf(v.x, 0.f); v.y = fmaxf(v.y, 0.f);
  v.z = fmaxf(v.z, 0.f); v.w = fmaxf(v.w, 0.f);
  ((float4*)x)[i] = v;
}

// out[i] = dot(h1[i][:], W2[:,0]) + b2   (pred second linear, fout==1)
__global__ __launch_bounds__(256) void pred_out_kernel(const float* __restrict__ h1,
    const float* __restrict__ W2, const float* __restrict__ b2,
    float* __restrict__ out, int n) {
  int wid  = (int)((blockIdx.x * blockDim.x + threadIdx.x) >> 5);
  int lane = (int)(threadIdx.x & 31);
  if (wid >= n) return;
  float4 h = ((const float4*)h1)[(size_t)wid * 32 + lane];
  float4 w = ((const float4*)W2)[lane];
  float p = h.x * w.x + h.y * w.y + h.z * w.z + h.w * w.w;
#pragma unroll
  for (int off = 16; off > 0; off >>= 1) p += __shfl_down(p, off, 32);
  if (lane == 0) out[wid] = p + b2[0];
}

} // namespace

// ---------------------------------------------------------------------------
// Input flattening assumption (jax tree_leaves; nested dict keys sorted):
//  0-3   enc_b : l1.W[1,128] l1.b[128] l2.W[128,128] l2.b[128]
//  4-7   enc_c
//  8-11  enc_x
//  12+14*l + (v2c?7:0) + {0:edge.W,1:edge.b,2:eps,3:l1.W,4:l1.b,5:l2.W,6:l2.b}
//  54-57 pred : l1.W l1.b l2.W[128,1] l2.b[1]
//  58 edge_index_c2v[2,E] 59 edge_attr_c2v[E] 60 edge_index_v2c 61 edge_attr_v2c
//  62 b[NC] 63 c[NV] 64 x_start[NV]
// ---------------------------------------------------------------------------
extern "C" void kernel_launch(void* const* d_in, const int* in_sizes, int n_in,
                              void* d_out, int out_size, void* d_ws, size_t ws_size,
                              hipStream_t stream) {
  (void)in_sizes; (void)n_in; (void)out_size; (void)ws_size;
  auto F = [&](int i) { return (const float*)d_in[i]; };
  auto I = [&](int i) { return (const int*)d_in[i]; };

  const int ENC_B = 0, ENC_C = 4, ENC_X = 8, PRED = 54;
  auto gineBase = [](int layer, bool v2c) { return 12 + layer * 14 + (v2c ? 7 : 0); };
  const int EI_C2V = 58, AT_C2V = 59, EI_V2C = 60, AT_V2C = 61;
  const int B_IDX = 62, C_IDX = 63, XS_IDX = 64;

  float* ws  = (float*)d_ws;
  float* xc  = ws; ws += (size_t)NC * HIDC;   // x_cons
  float* xv  = ws; ws += (size_t)NV * HIDC;   // x_vals
  float* agv = ws; ws += (size_t)NV * HIDC;   // aggr / z / h1 scratch (vals)
  float* agc = ws; ws += (size_t)NC * HIDC;   // aggr / z / h1 scratch (cons)
  float* hv  = ws; ws += (size_t)NV * HIDC;   // h_vals (pre-relu; kept for pred)
  float* hc  = ws; ws += (size_t)NC * HIDC;   // h_cons
  float* wpk = ws;                            // 16 packed 128x128 matrices (1 MB)

  auto wslot = [&](int s) { return wpk + (size_t)s * 16384; };
  auto pack  = [&](const float* W, int slot) {
    pack_w_kernel<<<32, 256, 0, stream>>>(W, (float2*)wslot(slot));
  };
  auto gemm = [&](const float* Z, int slot, const float* bias, float* OUT, int n, int flags) {
    gemm128_kernel<<<(n + 127) / 128, 256, 0, stream>>>(Z, wslot(slot), bias, OUT, n, flags);
  };
  auto zero = [&](float* p, size_t nf) {
    int n4 = (int)(nf / 4);
    zero_kernel<<<(n4 + 255) / 256, 256, 0, stream>>>((float4*)p, n4);
  };

  // Pack all GEMM weights into WMMA B-fragment layout (cheap; once per launch).
  pack(F(ENC_B + 2), 0);
  pack(F(ENC_C + 2), 1);
  pack(F(ENC_X + 2), 2);
  for (int l = 0; l < NLAY; ++l) {
    pack(F(gineBase(l, false) + 3), 3 + 4 * l);   // c2v l1.W
    pack(F(gineBase(l, false) + 5), 4 + 4 * l);   // c2v l2.W
    pack(F(gineBase(l, true)  + 3), 5 + 4 * l);   // v2c l1.W
    pack(F(gineBase(l, true)  + 5), 6 + 4 * l);   // v2c l2.W
  }
  pack(F(PRED + 0), 15);

  // Encoders: x_cons = mlp2(enc_b, b); x_vals = mlp2(enc_x, x0) + mlp2(enc_c, c)
  outer_relu_kernel<<<(NC + 7) / 8, 256, 0, stream>>>(F(B_IDX),  F(ENC_B + 0), F(ENC_B + 1), agc, NC);
  gemm(agc, 0, F(ENC_B + 3), xc, NC, 0);
  outer_relu_kernel<<<(NV + 7) / 8, 256, 0, stream>>>(F(XS_IDX), F(ENC_X + 0), F(ENC_X + 1), agv, NV);
  gemm(agv, 2, F(ENC_X + 3), xv, NV, 0);
  outer_relu_kernel<<<(NV + 7) / 8, 256, 0, stream>>>(F(C_IDX),  F(ENC_C + 0), F(ENC_C + 1), agv, NV);
  gemm(agv, 1, F(ENC_C + 3), xv, NV, GEMM_ADD);

  const int nv4 = NV * HIDC / 4, nc4 = NC * HIDC / 4;
  for (int l = 0; l < NLAY; ++l) {
    const int bc = gineBase(l, false);   // c2v params
    const int bv = gineBase(l, true);    // v2c params
    zero(agv, (size_t)NV * HIDC);
    zero(agc, (size_t)NC * HIDC);
    // Both directions read the layer-entry x_cons / x_vals.
    edge_msg_kernel<<<NE / 8, 256, 0, stream>>>(xc, I(EI_C2V), F(AT_C2V), F(bc + 0), F(bc + 1), agv, NE);
    edge_msg_kernel<<<NE / 8, 256, 0, stream>>>(xv, I(EI_V2C), F(AT_V2C), F(bv + 0), F(bv + 1), agc, NE);
    gine_pre_kernel<<<(nv4 + 255) / 256, 256, 0, stream>>>(agv, xv, F(bc + 2), nv4);
    gine_pre_kernel<<<(nc4 + 255) / 256, 256, 0, stream>>>(agc, xc, F(bv + 2), nc4);
    gemm(agv, 3 + 4 * l, F(bc + 4), agv, NV, GEMM_RELU);   // h1 (in place)
    gemm(agv, 4 + 4 * l, F(bc + 6), hv,  NV, 0);           // h_vals
    gemm(agc, 5 + 4 * l, F(bv + 4), agc, NC, GEMM_RELU);
    gemm(agc, 6 + 4 * l, F(bv + 6), hc,  NC, 0);           // h_cons
    relu_copy_kernel<<<(nv4 + 255) / 256, 256, 0, stream>>>(hv, xv, nv4);
    relu_copy_kernel<<<(nc4 + 255) / 256, 256, 0, stream>>>(hc, xc, nc4);
  }

  // Prediction head on hidden_vals (pre-relu h_vals of last layer).
  gemm(hv, 15, F(PRED + 1), agv, NV, GEMM_RELU);
  pred_out_kernel<<<(NV + 7) / 8, 256, 0, stream>>>(agv, F(PRED + 2), F(PRED + 3), (float*)d_out, NV);
}